// ISNEAttention_23622320128100
// MI455X (gfx1250) — compile-verified
//
#include <hip/hip_runtime.h>
#include <math.h>

typedef __attribute__((ext_vector_type(2))) float v2f;
typedef __attribute__((ext_vector_type(8))) float v8f;

#define D_DIM 64
#define H_DIM 64
#define LRELU_ALPHA 0.2f

// Monotone float <-> uint encoding so segment-max can use integer atomicMax.
__device__ __forceinline__ unsigned f2ord(float f) {
    unsigned u = __float_as_uint(f);
    return (u & 0x80000000u) ? ~u : (u | 0x80000000u);
}
__device__ __forceinline__ float ord2f(unsigned e) {
    return (e & 0x80000000u) ? __uint_as_float(e & 0x7fffffffu)
                             : __uint_as_float(~e);
}

// ---------------------------------------------------------------- init
__global__ void init_kernel(float* __restrict__ out, unsigned* __restrict__ m_ord,
                            float* __restrict__ denom, int N) {
    int i = blockIdx.x * blockDim.x + threadIdx.x;
    if (i < N * H_DIM) out[i] = 0.0f;
    if (i < N) {
        m_ord[i] = 0x007FFFFFu;   // f2ord(-inf)
        denom[i] = 0.0f;
    }
}

// --------------------------------------------- Wh = x @ W  (FP32 WMMA)
// One wave per 16 rows. 4 N-tiles of 16 cols, K swept in steps of 4 via
// V_WMMA_F32_16X16X4_F32. Epilogue fuses s1 = Wh·a1, s2 = Wh·a2 using
// 16-lane xor-shuffle reductions (stays within each half of the wave).
__global__ void gemm_wmma_kernel(const float* __restrict__ x,
                                 const float* __restrict__ W,
                                 const float* __restrict__ a_vec,
                                 float* __restrict__ Wh,
                                 float* __restrict__ s1,
                                 float* __restrict__ s2,
                                 int N) {
    const int lane = threadIdx.x & 31;
    const int half = lane >> 4;      // 0 or 1
    const int sub  = lane & 15;      // row (A) / col (B,C,D) within tile
    const int r0   = blockIdx.x * 16;
    if (r0 >= N) return;

    v8f acc[4];
    acc[0] = {}; acc[1] = {}; acc[2] = {}; acc[3] = {};

    const float* xrow = x + (size_t)(r0 + sub) * D_DIM;

    for (int k0 = 0; k0 < D_DIM; k0 += 4) {
        const int ka = k0 + 2 * half;           // this lane's K pair
        float2 av = *(const float2*)(xrow + ka); // A: (M=sub, K=ka, ka+1)
        v2f afrag; afrag.x = av.x; afrag.y = av.y;
#pragma unroll
        for (int t = 0; t < 4; ++t) {
            v2f bfrag;                           // B: (K=ka/ka+1, N=t*16+sub)
            bfrag.x = W[(size_t)ka * H_DIM + t * 16 + sub];
            bfrag.y = W[(size_t)(ka + 1) * H_DIM + t * 16 + sub];
            acc[t] = __builtin_amdgcn_wmma_f32_16x16x4_f32(
                false, afrag, false, bfrag, (short)0, acc[t], false, false);
        }
    }

    // Store Wh: c[v] holds (M = v + 8*half, N = t*16 + sub)
#pragma unroll
    for (int t = 0; t < 4; ++t)
#pragma unroll
        for (int v = 0; v < 8; ++v)
            Wh[(size_t)(r0 + v + 8 * half) * H_DIM + t * 16 + sub] = acc[t][v];

    // Fused per-row dots with a1 (= a[0:64]) and a2 (= a[64:128]).
#pragma unroll
    for (int v = 0; v < 8; ++v) {
        float p1 = 0.0f, p2 = 0.0f;
#pragma unroll
        for (int t = 0; t < 4; ++t) {
            float c = acc[t][v];
            p1 += c * a_vec[t * 16 + sub];
            p2 += c * a_vec[H_DIM + t * 16 + sub];
        }
#pragma unroll
        for (int off = 8; off >= 1; off >>= 1) {   // stays within 16-lane half
            p1 += __shfl_xor(p1, off, 32);
            p2 += __shfl_xor(p2, off, 32);
        }
        if (sub == 0) {
            s1[r0 + v + 8 * half] = p1;
            s2[r0 + v + 8 * half] = p2;
        }
    }
}

// ------------------------------------------------ pass 1: segment max
__global__ void edge_max_kernel(const int* __restrict__ src, const int* __restrict__ tgt,
                                const float* __restrict__ s1, const float* __restrict__ s2,
                                unsigned* __restrict__ m_ord, int E) {
    int j = blockIdx.x * blockDim.x + threadIdx.x;
    if (j >= E) return;
    int s = src[j], t = tgt[j];
    float e = s1[s] + s2[t];
    e = e > 0.0f ? e : LRELU_ALPHA * e;
    atomicMax(&m_ord[s], f2ord(e));
}

// --------------------------------------- pass 2: exp + segment sum
__global__ void edge_expsum_kernel(const int* __restrict__ src, const int* __restrict__ tgt,
                                   const float* __restrict__ s1, const float* __restrict__ s2,
                                   const unsigned* __restrict__ m_ord,
                                   float* __restrict__ denom, float* __restrict__ ex, int E) {
    int j = blockIdx.x * blockDim.x + threadIdx.x;
    if (j >= E) return;
    int s = src[j], t = tgt[j];
    float e = s1[s] + s2[t];
    e = e > 0.0f ? e : LRELU_ALPHA * e;
    float v = __expf(e - ord2f(m_ord[s]));
    ex[j] = v;
    atomicAdd(&denom[s], v);
}

// ----------------------- pass 3: out[src] += att * Wh[tgt] (16 lanes/edge)
__global__ void edge_scatter_kernel(const int* __restrict__ src, const int* __restrict__ tgt,
                                    const float* __restrict__ ex, const float* __restrict__ denom,
                                    const float* __restrict__ Wh, float* __restrict__ out, int E) {
    int tid = blockIdx.x * blockDim.x + threadIdx.x;
    int j = tid >> 4;
    int g = tid & 15;
    if (j >= E) return;
    int s = src[j], t = tgt[j];
    float att = ex[j] / denom[s];
    const float4 wh = *(const float4*)(Wh + (size_t)t * H_DIM + g * 4);
    float* o = out + (size_t)s * H_DIM + g * 4;
    atomicAdd(o + 0, wh.x * att);
    atomicAdd(o + 1, wh.y * att);
    atomicAdd(o + 2, wh.z * att);
    atomicAdd(o + 3, wh.w * att);
}

// ------------------------------------------------------------ launch
extern "C" void kernel_launch(void* const* d_in, const int* in_sizes, int n_in,
                              void* d_out, int out_size, void* d_ws, size_t ws_size,
                              hipStream_t stream) {
    const float* x = (const float*)d_in[0];
    const float* W = (const float*)d_in[1];
    const float* a = (const float*)d_in[2];
    const int* edge_index = (const int*)d_in[3];

    const int N = in_sizes[0] / D_DIM;
    const int E = in_sizes[3] / 2;
    const int* src = edge_index;       // edge_index[0, :]
    const int* tgt = edge_index + E;   // edge_index[1, :]
    float* out = (float*)d_out;

    // Workspace carve-up (floats): Wh | s1 | s2 | m_ord | denom | ex
    float* ws = (float*)d_ws;
    float* Wh = ws;                     ws += (size_t)N * H_DIM;
    float* s1 = ws;                     ws += N;
    float* s2 = ws;                     ws += N;
    unsigned* m_ord = (unsigned*)ws;    ws += N;
    float* denom = ws;                  ws += N;
    float* ex = ws;                     ws += E;

    const int total = N * H_DIM;
    init_kernel<<<(total + 255) / 256, 256, 0, stream>>>(out, m_ord, denom, N);
    gemm_wmma_kernel<<<(N + 15) / 16, 32, 0, stream>>>(x, W, a, Wh, s1, s2, N);
    edge_max_kernel<<<(E + 255) / 256, 256, 0, stream>>>(src, tgt, s1, s2, m_ord, E);
    edge_expsum_kernel<<<(E + 255) / 256, 256, 0, stream>>>(src, tgt, s1, s2, m_ord, denom, ex, E);
    edge_scatter_kernel<<<((size_t)E * 16 + 255) / 256, 256, 0, stream>>>(src, tgt, ex, denom, Wh, out, E);
}